// SpikingReservoirLoaded_43215960932429
// MI455X (gfx1250) — compile-verified
//
#include <hip/hip_runtime.h>
#include <stdint.h>

// ---------------------------------------------------------------------------
// Spiking reservoir, B=32, T=1000, N=1000, fp32 end-to-end.
// One kernel launch per timestep (device-wide sync, graph-capturable).
// Per-step GEMM S(32x1000)@W(1000x1000) on the CDNA5 matrix pipe via
// V_WMMA_F32_16X16X4_F32 (exact fp32: spikes are exactly 0/1, W stays fp32).
// W (4MB) is L2-resident (192MB); the 256MB of records stream out with
// non-temporal stores so they don't evict the reused working set.
// ---------------------------------------------------------------------------

typedef float v2f __attribute__((ext_vector_type(2)));
typedef float v8f __attribute__((ext_vector_type(8)));

static constexpr int   kB      = 32;
static constexpr int   kN      = 1000;
static constexpr int   kT      = 1000;
static constexpr float kThresh = 1.0f;
static constexpr float kBeta   = 0.9f;

// --------------------------- init: zero state ------------------------------
__global__ void init_state(float* __restrict__ V,
                           float* __restrict__ S0,
                           float* __restrict__ S1,
                           unsigned int* __restrict__ cnt) {
    int i = blockIdx.x * blockDim.x + threadIdx.x;
    int n = kB * kN;
    if (i < n) { V[i] = 0.0f; S0[i] = 0.0f; S1[i] = 0.0f; }
    if (i == 0) *cnt = 0u;
}

// Deterministic stand-in for jax.random.uniform(key(42), (B,1)): host-side
// Threefry cannot be reproduced on-device without host constants. Fixed hash
// per batch index -> uniform [0,1), identical on every call (graph-safe).
__device__ __forceinline__ float uniform_for_batch(uint32_t b) {
    uint32_t x = b * 747796405u + 2891336453u;
    x = ((x >> ((x >> 28u) + 4u)) ^ x) * 277803737u;
    x = (x >> 22u) ^ x;
    return (float)(x >> 8) * (1.0f / 16777216.0f);
}

// --------------------------- one timestep ----------------------------------
// grid.x = 63 column tiles (N padded to 1008), block = 64 threads = 2 waves.
// Wave w owns the 16x16 output tile (rows m0=16*w .. m0+15, cols n0..n0+15).
__global__ __launch_bounds__(64)
void reservoir_step(const float* __restrict__ W,
                    const float* __restrict__ Sprev,
                    float* __restrict__ Snext,
                    float* __restrict__ V,
                    float* __restrict__ spike_rec,   // (T,B,N) at d_out+1
                    float* __restrict__ mem_rec,     // (T,B,N) after spikes
                    unsigned int* __restrict__ spike_cnt,
                    int t)
{
    const int lane = threadIdx.x & 31;
    const int wave = threadIdx.x >> 5;     // 0/1 -> M-tile (batch rows)
    const int m0   = wave * 16;
    const int n0   = blockIdx.x * 16;
    const int row  = lane & 15;
    const int hi   = lane >> 4;            // half-wave -> K striping

    // Column masking done arithmetically: WMMA requires EXEC all-ones, so no
    // divergent branches around the matrix op. OOB columns read a clamped
    // in-bounds address and are multiplied by 0.
    const int   col   = n0 + row;
    const int   ccol  = (col < kN) ? col : (kN - 1);
    const float bmask = (col < kN) ? 1.0f : 0.0f;

    // A-tile (16x4 fp32, ISA 7.12.2): lane holds M=row, VGPR0=K(2*hi),
    // VGPR1=K(2*hi+1). B-tile (4x16): same K striping, N=row across lanes.
    v8f c = {};
    const float* arow = Sprev + (size_t)(m0 + row) * kN;

    // K = 1000 = 50 outer iterations x 5 WMMAs (K=4 each). One clamped,
    // unconditional WGP-scope prefetch per 20-K body keeps the W stream
    // flowing L2 -> WGP$ without any branches in the hot loop.
    for (int k = 0; k < kN; k += 20) {
        const int kp = (k + 40 < kN) ? (k + 40) : (kN - 20);
        __builtin_prefetch(W + (size_t)(kp + 2 * hi) * kN + ccol, 0, 3);
        #pragma unroll
        for (int kk = 0; kk < 20; kk += 4) {
            const int ka = k + kk + 2 * hi;
            v2f a = { arow[ka], arow[ka + 1] };
            const float* wb = W + (size_t)ka * kN + ccol;
            v2f b = { wb[0] * bmask, wb[kN] * bmask };
            // (neg_a, A, neg_b, B, c_mod, C, reuse_a, reuse_b) -> f32 WMMA
            c = __builtin_amdgcn_wmma_f32_16x16x4_f32(false, a, false, b,
                                                      (short)0, c,
                                                      false, false);
        }
    }

    // Epilogue. C/D layout: VGPR i -> M = i + 8*hi, N = n0 + row.
    unsigned int my_spk = 0;
    if (col < kN) {
        #pragma unroll
        for (int i = 0; i < 8; ++i) {
            const int   m    = m0 + i + 8 * hi;               // batch 0..31
            const float iext = (t == 0) ? (2.0f * kThresh)
                                        : uniform_for_batch((uint32_t)m);
            const size_t idx = (size_t)m * kN + col;
            const float v  = kBeta * V[idx] + iext + c[i];
            const float s  = (v >= kThresh) ? 1.0f : 0.0f;
            const float vr = v * (1.0f - s);                  // reset-to-zero
            V[idx]     = vr;           // re-read next step: keep in L2 (RT)
            Snext[idx] = s;            // re-read next step: keep in L2 (RT)
            const size_t r = (size_t)t * (size_t)(kB * kN) + idx;
            __builtin_nontemporal_store(s,  &spike_rec[r]);   // write-once
            __builtin_nontemporal_store(vr, &mem_rec[r]);     // stream (NT)
            my_spk += (unsigned int)s;
        }
    }
    // Exact integer spike count: wave32 butterfly + one atomic per wave.
    unsigned int tot = my_spk;
    #pragma unroll
    for (int off = 16; off > 0; off >>= 1)
        tot += __shfl_xor(tot, off, 32);
    if (lane == 0) atomicAdd(spike_cnt, tot);
}

// --------------------------- finalize --------------------------------------
__global__ void finalize_rate(const unsigned int* __restrict__ cnt,
                              float* __restrict__ out) {
    if (threadIdx.x == 0 && blockIdx.x == 0)
        out[0] = (float)(*cnt) / (float)((long long)kT * kB * kN);
}

// ---------------------------------------------------------------------------
extern "C" void kernel_launch(void* const* d_in, const int* in_sizes, int n_in,
                              void* d_out, int out_size, void* d_ws, size_t ws_size,
                              hipStream_t stream) {
    (void)in_sizes; (void)n_in; (void)out_size; (void)ws_size;

    // setup_inputs order: x (unused by the reference math), W.
    const float* W   = (const float*)d_in[1];
    float*       out = (float*)d_out;

    // Workspace: V (B*N f32) | S0 | S1 | spike counter  (~384 KB).
    float* V  = (float*)d_ws;
    float* S0 = V  + (size_t)kB * kN;
    float* S1 = S0 + (size_t)kB * kN;
    unsigned int* cnt = (unsigned int*)(S1 + (size_t)kB * kN);

    {
        const int n = kB * kN;
        init_state<<<(n + 255) / 256, 256, 0, stream>>>(V, S0, S1, cnt);
    }

    // d_out layout (tuple, flat, return order): [avg | spikes(T,B,N) | mem(T,B,N)]
    float* spike_rec = out + 1;
    float* mem_rec   = spike_rec + (size_t)kT * kB * kN;

    const dim3 grid((kN + 15) / 16);   // 63 column tiles
    const dim3 block(64);              // 2 waves (wave32)
    for (int t = 0; t < kT; ++t) {
        const float* Sp = (t & 1) ? S1 : S0;   // double-buffered spikes:
        float*       Sn = (t & 1) ? S0 : S1;   // no intra-step RAW races
        reservoir_step<<<grid, block, 0, stream>>>(W, Sp, Sn, V,
                                                   spike_rec, mem_rec, cnt, t);
    }
    finalize_rate<<<1, 1, 0, stream>>>(cnt, out);
}